// PillarEncoder_83537113907657
// MI455X (gfx1250) — compile-verified
//
#include <hip/hip_runtime.h>

// PointPillars pillar encoder for MI455X (gfx1250, wave32).
// Bandwidth-bound (~270MB moved vs 2.4 GFLOP) -> keep fp32 exactness and use
// V_WMMA_F32_16X16X4_F32 for the (32x9)@(9x64) per-pillar GEMM (K padded to 12).

typedef __attribute__((ext_vector_type(2))) float v2f;
typedef __attribute__((ext_vector_type(8))) float v8f;

#define X_L   432
#define Y_L   496
#define C_OUT 64
#define PPTS  32
#define BN_EPS 1e-3f

__device__ __forceinline__ float wave_reduce_add(float v) {
#pragma unroll
  for (int m = 16; m >= 1; m >>= 1) v += __shfl_xor(v, m, 32);
  return v;
}

// A-matrix 16x4 f32 layout (ISA 7.12.2): lanes 0-15 row M=lane hold K=0 (v0),
// K=1 (v1); lanes 16-31 same rows hold K=2 (v0), K=3 (v1).
// Source features live in lane `base + (lane&15)`'s registers -> shfl gather.
template<int KS>
__device__ __forceinline__ v2f a_frag(const float (&f)[12], int base, int lane) {
  const int src = base + (lane & 15);
  const float lo0 = __shfl(f[KS * 4 + 0], src, 32);
  const float lo1 = __shfl(f[KS * 4 + 1], src, 32);
  const float hi0 = __shfl(f[KS * 4 + 2], src, 32);
  const float hi1 = __shfl(f[KS * 4 + 3], src, 32);
  const bool hi = lane >= 16;
  v2f a;
  a[0] = hi ? hi0 : lo0;
  a[1] = hi ? hi1 : lo1;
  return a;
}

// B-matrix 4x16 f32: lane (0-15) = column N; lanes 0-15 hold K=0,1 in v0,v1;
// lanes 16-31 hold K=2,3.  B[k][n] = W[n][k] (W is C_OUT x 9 row-major).
__device__ __forceinline__ v2f b_frag(const float* __restrict__ W, int t, int ks,
                                      int lane) {
  const int c = t * 16 + (lane & 15);
  const int k = ks * 4 + ((lane >> 4) << 1);
  v2f b;
  b[0] = (k + 0 < 9) ? W[c * 9 + k + 0] : 0.0f;
  b[1] = (k + 1 < 9) ? W[c * 9 + k + 1] : 0.0f;
  return b;
}

__global__ __launch_bounds__(256) void pillar_zero_kernel(float4* __restrict__ out,
                                                          long n4, long ntail,
                                                          float* __restrict__ outf) {
  const float4 z = make_float4(0.f, 0.f, 0.f, 0.f);
  long i = (long)blockIdx.x * blockDim.x + threadIdx.x;
  const long stride = (long)gridDim.x * blockDim.x;
  for (; i < n4; i += stride) out[i] = z;
  if (blockIdx.x == 0 && threadIdx.x < ntail) outf[n4 * 4 + threadIdx.x] = 0.f;
}

__global__ __launch_bounds__(256) void pillar_encoder_kernel(
    const float* __restrict__ pillars, const int* __restrict__ coors,
    const int* __restrict__ nums, const float* __restrict__ W,
    const float* __restrict__ gamma, const float* __restrict__ beta,
    const float* __restrict__ rmean, const float* __restrict__ rvar,
    float* __restrict__ out, int M) {
  const int lane = threadIdx.x & 31;
  const int wave = blockIdx.x * (blockDim.x >> 5) + (threadIdx.x >> 5);
  const int nwaves = gridDim.x * (blockDim.x >> 5);

  // Loop-invariant per-lane state: B fragments (4 col tiles x 3 K-steps) and
  // fused BN scale/bias for channel c = t*16 + (lane&15).
  v2f bf[4][3];
  float sc[4], bi[4];
#pragma unroll
  for (int t = 0; t < 4; ++t) {
#pragma unroll
    for (int ks = 0; ks < 3; ++ks) bf[t][ks] = b_frag(W, t, ks, lane);
    const int c = t * 16 + (lane & 15);
    const float s = gamma[c] * rsqrtf(rvar[c] + BN_EPS);
    sc[t] = s;
    bi[t] = beta[c] - rmean[c] * s;
  }

  for (int m = wave; m < M; m += nwaves) {
    const int cb = coors[m * 4 + 0];
    const int cy = coors[m * 4 + 2];
    const int cx = coors[m * 4 + 3];
    const int num = nums[m];

    // lane = point index within pillar; 16B coalesced load.
    const float4 pt = reinterpret_cast<const float4*>(pillars)[(long)m * PPTS + lane];
    const float mask = (lane < num) ? 1.0f : 0.0f;
    const float inv = 1.0f / fmaxf((float)num, 1.0f);
    const float mx = wave_reduce_add(pt.x * mask) * inv;
    const float my = wave_reduce_add(pt.y * mask) * inv;
    const float mz = wave_reduce_add(pt.z * mask) * inv;
    const float xc = (float)cx * 0.16f + 0.08f;            // VX/2 + X0
    const float yc = (float)cy * 0.16f + (0.08f - 39.68f); // VY/2 + Y0

    float f[12];
    f[0] = pt.x * mask;
    f[1] = pt.y * mask;
    f[2] = pt.z * mask;
    f[3] = pt.w * mask;
    f[4] = (pt.x - mx) * mask;
    f[5] = (pt.y - my) * mask;
    f[6] = (pt.z - mz) * mask;
    f[7] = (pt.x - xc) * mask;
    f[8] = (pt.y - yc) * mask;
    f[9] = 0.f; f[10] = 0.f; f[11] = 0.f;

    // A fragments: row tile 0 = points 0..15, row tile 1 = points 16..31.
    v2f a0[3], a1[3];
    a0[0] = a_frag<0>(f, 0, lane);
    a0[1] = a_frag<1>(f, 0, lane);
    a0[2] = a_frag<2>(f, 0, lane);
    a1[0] = a_frag<0>(f, 16, lane);
    a1[1] = a_frag<1>(f, 16, lane);
    a1[2] = a_frag<2>(f, 16, lane);

    const bool valid =
        (cx >= 0) && (cx < X_L) && (cy >= 0) && (cy < Y_L) && (cb >= 0) && (cb < 4);

#pragma unroll
    for (int t = 0; t < 4; ++t) {
      v8f acc0 = {};
      v8f acc1 = {};
#pragma unroll
      for (int ks = 0; ks < 3; ++ks) {
        acc0 = __builtin_amdgcn_wmma_f32_16x16x4_f32(false, a0[ks], false, bf[t][ks],
                                                     (short)0, acc0, false, false);
        acc1 = __builtin_amdgcn_wmma_f32_16x16x4_f32(false, a1[ks], false, bf[t][ks],
                                                     (short)0, acc1, false, false);
      }
      // BN + ReLU + max over this lane's 16 rows (8 from each row tile).
      float vmax = 0.0f;
#pragma unroll
      for (int r = 0; r < 8; ++r) {
        vmax = fmaxf(vmax, fmaxf(acc0[r] * sc[t] + bi[t], 0.0f));
        vmax = fmaxf(vmax, fmaxf(acc1[r] * sc[t] + bi[t], 0.0f));
      }
      // Combine the two lane halves holding the same column (channel).
      vmax = fmaxf(vmax, __shfl_xor(vmax, 16, 32));
      if (valid && lane < 16) {
        const int c = t * 16 + lane;
        out[(((long)cb * C_OUT + c) * Y_L + cy) * X_L + cx] = vmax;
      }
    }
  }
}

extern "C" void kernel_launch(void* const* d_in, const int* in_sizes, int n_in,
                              void* d_out, int out_size, void* d_ws, size_t ws_size,
                              hipStream_t stream) {
  const float* pillars = (const float*)d_in[0];
  const int* coors = (const int*)d_in[1];
  const int* nums = (const int*)d_in[2];
  const float* W = (const float*)d_in[3];
  const float* gamma = (const float*)d_in[4];
  const float* beta = (const float*)d_in[5];
  const float* rmean = (const float*)d_in[6];
  const float* rvar = (const float*)d_in[7];
  float* out = (float*)d_out;
  const int M = in_sizes[2];  // number of pillars (nums has one entry each)

  const long n4 = (long)out_size / 4;
  const long ntail = (long)out_size - n4 * 4;
  // 219 MB canvas zero-fill: pure-bandwidth grid-stride float4 stores.
  pillar_zero_kernel<<<4096, 256, 0, stream>>>((float4*)d_out, n4, ntail, out);
  // One wave per pillar, 8 pillars per wave via grid-stride (8000 waves).
  pillar_encoder_kernel<<<1000, 256, 0, stream>>>(pillars, coors, nums, W, gamma,
                                                  beta, rmean, rvar, out, M);
}